// RasSampler_34900904247421
// MI455X (gfx1250) — compile-verified
//
#include <hip/hip_runtime.h>
#include <hip/hip_bf16.h>

// RasSampler for MI455X (gfx1250): one 256-thread (8×wave32) block per row.
// Logits row (26.25 KB) staged HBM->LDS once via the Tensor Data Mover, then
// all passes run out of LDS. HBM traffic ~= 110 MB total -> ~4.7 us floor.

#define V_DIM     6562
#define H_DIM     200
#define EOS_TOK   6561
#define WIN_SZ    10
#define TOPK      25
#define TOPP      0.8f
#define MAX_TRIALS 100
#define NTHREADS  256
#define NEG_INF   (-3.402823e38f)

typedef unsigned int u32x4 __attribute__((ext_vector_type(4)));
typedef int          i32x8 __attribute__((ext_vector_type(8)));
typedef int          i32x4 __attribute__((ext_vector_type(4)));

#if defined(__has_builtin)
#if __has_builtin(__builtin_amdgcn_tensor_load_to_lds) && __has_builtin(__builtin_amdgcn_s_wait_tensorcnt)
#define USE_TDM 1
#endif
#endif

__device__ __forceinline__ unsigned pcg_hash(unsigned x) {
  x = x * 747796405u + 2891336453u;
  unsigned w = ((x >> ((x >> 28u) + 4u)) ^ x) * 277803737u;
  return (w >> 22u) ^ w;
}

__device__ __forceinline__ float gumbel_from(unsigned h) {
  float u = ((float)(h >> 8) + 0.5f) * (1.0f / 16777216.0f); // (0,1)
  return -__logf(-__logf(u));
}

// ---- block-wide reductions (256 threads = 8 wave32 waves) ----
__device__ __forceinline__ float block_max(float v, float* s_rv, int tid) {
  __syncthreads();
  s_rv[tid] = v;
  __syncthreads();
  for (int s = NTHREADS / 2; s > 0; s >>= 1) {
    if (tid < s) s_rv[tid] = fmaxf(s_rv[tid], s_rv[tid + s]);
    __syncthreads();
  }
  return s_rv[0];
}

__device__ __forceinline__ float block_sum(float v, float* s_rv, int tid) {
  __syncthreads();
  s_rv[tid] = v;
  __syncthreads();
  for (int s = NTHREADS / 2; s > 0; s >>= 1) {
    if (tid < s) s_rv[tid] += s_rv[tid + s];
    __syncthreads();
  }
  return s_rv[0];
}

// argmax -> result in s_rv[0]/s_ri[0] (all threads may read after return)
__device__ __forceinline__ void block_argmax(float v, int idx, float* s_rv, int* s_ri, int tid) {
  __syncthreads();
  s_rv[tid] = v; s_ri[tid] = idx;
  __syncthreads();
  for (int s = NTHREADS / 2; s > 0; s >>= 1) {
    if (tid < s) {
      float ov = s_rv[tid + s]; int oi = s_ri[tid + s];
      if (ov > s_rv[tid] || (ov == s_rv[tid] && oi < s_ri[tid])) {
        s_rv[tid] = ov; s_ri[tid] = oi;
      }
    }
    __syncthreads();
  }
}

// categorical over softmax(logits) via Gumbel-max on the LDS-resident row
__device__ __forceinline__ int gumbel_argmax_full(const float* s_logits, float* s_rv,
                                                  int* s_ri, int tid, unsigned seed) {
  float bv = NEG_INF; int bi = 0;
  for (int i = tid; i < V_DIM; i += NTHREADS) {
    float sc = s_logits[i] + gumbel_from(pcg_hash(seed ^ ((unsigned)i * 0x9E3779B1u)));
    if (sc > bv) { bv = sc; bi = i; }
  }
  block_argmax(bv, bi, s_rv, s_ri, tid);
  return s_ri[0];
}

__global__ __launch_bounds__(NTHREADS)
void ras_sampler_kernel(const float* __restrict__ logits,
                        const int*   __restrict__ decoded_tokens,
                        const int*   __restrict__ decoded_lens,
                        const int*   __restrict__ min_tokens,
                        float*       __restrict__ out) {
  __shared__ float s_logits[V_DIM];
  __shared__ float s_rv[NTHREADS];
  __shared__ int   s_ri[NTHREADS];
  __shared__ float s_topv[TOPK];
  __shared__ int   s_topi[TOPK];
  __shared__ int   s_flags[4];   // [0]=top_id  [1]=need_random  [2]=need_retry

  const int row = blockIdx.x;
  const int tid = threadIdx.x;
  const int len    = decoded_lens[row];
  const int mintok = min_tokens[row];
  const float* grow = logits + (size_t)row * V_DIM;

  if (tid == 0)
    __builtin_prefetch(decoded_tokens + (size_t)row * H_DIM, 0, 3); // global_prefetch_b8

  // ---- stage logits row into LDS via Tensor Data Mover ----
#ifdef USE_TDM
  if (tid < 32) { // wave 0 only issues the DMA; TDM ignores EXEC
    unsigned lds_addr = (unsigned)(size_t)(void*)&s_logits[0]; // LDS byte offset (addr[31:0])
    unsigned long long ga = (unsigned long long)(size_t)(const void*)grow;
    u32x4 g0; i32x8 g1;
    i32x4 g2 = {0, 0, 0, 0};
    i32x4 g3 = {0, 0, 0, 0};
    i32x8 g4 = {0, 0, 0, 0, 0, 0, 0, 0};
    // D# group0: count=1, lds_addr, global_addr[56:0], type=2
    g0[0] = 1u;
    g0[1] = lds_addr;
    g0[2] = (unsigned)(ga & 0xFFFFFFFFull);
    g0[3] = (unsigned)((ga >> 32) & 0x1FFFFFFull) | (2u << 30);
    // D# group1: data_size=4B, tensor 6562x1, tile 6562x1, dim0_stride=6562
    g1[0] = (int)(2u << 16);                              // data_size=2 (4 bytes)
    g1[1] = (int)((V_DIM & 0xFFFF) << 16);                // tensor_dim0[15:0]
    g1[2] = (int)(((V_DIM >> 16) & 0xFFFF) | (1 << 16));  // dim0[31:16] | tensor_dim1=1
    g1[3] = (int)((V_DIM & 0xFFFF) << 16);                // tile_dim0=V
    g1[4] = 1;                                            // tile_dim1=1, tile_dim2=0
    g1[5] = V_DIM;                                        // tensor_dim0_stride[31:0]
    g1[6] = 0; g1[7] = 0;
    __builtin_amdgcn_tensor_load_to_lds(g0, g1, g2, g3, g4, 0);
    __builtin_amdgcn_s_wait_tensorcnt(0);
  }
#else
  for (int i = tid; i < V_DIM; i += NTHREADS) s_logits[i] = grow[i];
#endif
  __syncthreads();

  // 1) forbid EOS for empty histories
  if (tid == 0 && len == 0) s_logits[EOS_TOK] = NEG_INF;
  __syncthreads();

  // 2) softmax stats from LDS
  float lm = NEG_INF;
  for (int i = tid; i < V_DIM; i += NTHREADS) lm = fmaxf(lm, s_logits[i]);
  const float m = block_max(lm, s_rv, tid);

  float ls = 0.0f;
  for (int i = tid; i < V_DIM; i += NTHREADS) ls += __expf(s_logits[i] - m);
  const float Z = block_sum(ls, s_rv, tid);

  // 3) top-25 selection (iterative argmax; claim via -inf, restore after)
  for (int k = 0; k < TOPK; ++k) {
    float bv = NEG_INF; int bi = 0;
    for (int i = tid; i < V_DIM; i += NTHREADS) {
      float v = s_logits[i];
      if (v > bv) { bv = v; bi = i; }
    }
    block_argmax(bv, bi, s_rv, s_ri, tid);
    if (tid == 0) {
      s_topv[k] = s_rv[0];
      s_topi[k] = s_ri[0];
      s_logits[s_ri[0]] = NEG_INF;
    }
    __syncthreads();
  }
  if (tid < TOPK) s_logits[s_topi[tid]] = s_topv[tid];  // restore full distribution
  __syncthreads();

  // 4) nucleus (top-p & top-k) sample + repetition window (cheap: thread 0)
  if (tid == 0) {
    unsigned seed0 = 0xC0FFEE17u ^ ((unsigned)row * 0x9E3779B9u);
    float cum = 0.0f, best = NEG_INF; int pos = 0;
    for (int k = 0; k < TOPK; ++k) {
      float p = __expf(s_topv[k] - m) / Z;
      cum += p;
      bool keep = (k == 0) || (cum <= TOPP);
      if (keep) {
        float sc = __logf(p) + gumbel_from(pcg_hash(seed0 + (unsigned)k));
        if (sc > best) { best = sc; pos = k; }
      }
    }
    int top_id = s_topi[pos];

    int lo = len - WIN_SZ; if (lo < 0) lo = 0;
    int cnt = 0;
    const int* drow = decoded_tokens + (size_t)row * H_DIM;
    for (int p2 = lo; p2 < len; ++p2) cnt += (drow[p2] == top_id) ? 1 : 0;
    int recent = (len < WIN_SZ) ? len : WIN_SZ;
    if (recent < 1) recent = 1;
    s_flags[0] = top_id;
    s_flags[1] = ((10 * cnt >= recent) && (len > 0)) ? 1 : 0;  // rep_rate >= 0.1
  }
  __syncthreads();

  // repetition-aware resample over the full distribution
  if (s_flags[1]) {  // block-uniform
    int rid = gumbel_argmax_full(s_logits, s_rv, s_ri, tid,
                                 0xB5297A4Du ^ ((unsigned)row * 0x68E31DA4u));
    if (tid == 0) s_flags[0] = rid;
  }
  __syncthreads();

  // 5) EOS retry loop (up to MAX_TRIALS) when history shorter than min_tokens
  if (tid == 0)
    s_flags[2] = ((len < mintok) && (s_flags[0] == EOS_TOK)) ? 1 : 0;
  __syncthreads();

  for (int t = 0; t < MAX_TRIALS; ++t) {
    if (!s_flags[2]) break;  // block-uniform
    int nid = gumbel_argmax_full(s_logits, s_rv, s_ri, tid,
                                 ((unsigned)(t + 1000) * 0x2545F491u) ^ (unsigned)row);
    if (tid == 0) {
      s_flags[0] = nid;
      s_flags[2] = ((len < mintok) && (nid == EOS_TOK)) ? 1 : 0;
    }
    __syncthreads();
  }

  if (tid == 0) {
    int top = s_flags[0];
    if (s_flags[2]) top = s_topi[1];  // fallback: second-best token
    out[row] = (float)top;
  }
}

extern "C" void kernel_launch(void* const* d_in, const int* in_sizes, int n_in,
                              void* d_out, int out_size, void* d_ws, size_t ws_size,
                              hipStream_t stream) {
  (void)n_in; (void)d_ws; (void)ws_size; (void)out_size;
  const float* logits         = (const float*)d_in[0];
  const int*   decoded_tokens = (const int*)d_in[1];
  const int*   decoded_lens   = (const int*)d_in[2];
  const int*   min_tokens     = (const int*)d_in[3];
  float*       out            = (float*)d_out;
  const int B = in_sizes[2];  // one block per row
  ras_sampler_kernel<<<dim3(B), dim3(NTHREADS), 0, stream>>>(
      logits, decoded_tokens, decoded_lens, min_tokens, out);
}